// GIN_17162689314903
// MI455X (gfx1250) — compile-verified
//
#include <hip/hip_runtime.h>
#include <hip/hip_bf16.h>

typedef __attribute__((ext_vector_type(2))) float v2f;
typedef __attribute__((ext_vector_type(8))) float v8f;

#define DIM 128
#define NLAYERS 5

// ---------------------------------------------------------------- zero
__global__ void gin_zero_f32(float* __restrict__ p, long long n) {
    long long i = (long long)blockIdx.x * blockDim.x + threadIdx.x;
    long long stride = (long long)gridDim.x * blockDim.x;
    for (; i < n; i += stride) p[i] = 0.0f;
}

// ---------------------------------------------------------------- scatter-add aggregation
// 32 lanes per edge, float4 per lane: one wave moves one edge's full 128-f32 row (512B coalesced).
__global__ void gin_scatter_agg(const float* __restrict__ h, const int* __restrict__ src,
                                const int* __restrict__ dst, float* __restrict__ agg, int nE) {
    long long idx = (long long)blockIdx.x * blockDim.x + threadIdx.x;
    int e = (int)(idx >> 5);
    if (e >= nE) return;
    int g = (int)(idx & 31) * 4;
    int s = src[e], d = dst[e];
    float4 v = *(const float4*)(h + (long long)s * DIM + g);
    float* p = agg + (long long)d * DIM + g;
    atomicAdd(p + 0, v.x);
    atomicAdd(p + 1, v.y);
    atomicAdd(p + 2, v.z);
    atomicAdd(p + 3, v.w);
}

// ---------------------------------------------------------------- WMMA f32 GEMM:  C[nrows x 128] = op(A) @ B + bias
// mode 0: op(A) = A0 + A1            (GIN: h + agg)
// mode 1: op(A) = relu(A0*scale+shift)  (fused BN+ReLU epilogue of previous GEMM)
// Block = 128 threads (4 waves); each wave computes a 16x128 strip via V_WMMA_F32_16X16X4_F32.
__global__ __launch_bounds__(128) void gin_gemm128(
    const float* __restrict__ A0, const float* __restrict__ A1,
    const float* __restrict__ scale, const float* __restrict__ shift,
    const float* __restrict__ B, const float* __restrict__ bias,
    float* __restrict__ C, int nrows, int mode) {
    __shared__ float lA[64][132];   // pad 4: frag bank = (4*row + col) % 64 -> conflict-free
    __shared__ float lB[32][136];   // pad 8: half-wave K-groups land in disjoint bank windows

    const int t = threadIdx.x;
    const int rowbase = blockIdx.x * 64;

    // ---- stage A tile (64 rows x 128 cols), fused pre-op; wave-uniform row => uniform bounds branch
    #pragma unroll
    for (int it = 0; it < 16; ++it) {
        int idx = it * 128 + t;          // float4 index, 0..2047
        int row = idx >> 5;              // 32 float4 per row
        int c4  = (idx & 31) * 4;
        int grow = rowbase + row;
        float4 v = make_float4(0.f, 0.f, 0.f, 0.f);
        if (grow < nrows) {
            v = *(const float4*)(A0 + (long long)grow * DIM + c4);
            if (mode == 0) {
                float4 w = *(const float4*)(A1 + (long long)grow * DIM + c4);
                v.x += w.x; v.y += w.y; v.z += w.z; v.w += w.w;
            } else {
                v.x = fmaxf(v.x * scale[c4 + 0] + shift[c4 + 0], 0.f);
                v.y = fmaxf(v.y * scale[c4 + 1] + shift[c4 + 1], 0.f);
                v.z = fmaxf(v.z * scale[c4 + 2] + shift[c4 + 2], 0.f);
                v.w = fmaxf(v.w * scale[c4 + 3] + shift[c4 + 3], 0.f);
            }
        }
        lA[row][c4 + 0] = v.x; lA[row][c4 + 1] = v.y;
        lA[row][c4 + 2] = v.z; lA[row][c4 + 3] = v.w;
    }

    const int lane = t & 31;
    const int wv   = t >> 5;      // wave id 0..3 -> rows wv*16..wv*16+15
    const int half = lane >> 4;   // ISA 7.12.2: lanes>=16 hold K+2,K+3
    const int l15  = lane & 15;

    v8f acc[8];
    const v8f vzero = {0.f, 0.f, 0.f, 0.f, 0.f, 0.f, 0.f, 0.f};
    #pragma unroll
    for (int ct = 0; ct < 8; ++ct) acc[ct] = vzero;

    for (int kb = 0; kb < 4; ++kb) {
        __syncthreads();
        // stage B chunk: rows [kb*32, kb*32+32) x 128 cols
        #pragma unroll
        for (int it = 0; it < 8; ++it) {
            int idx = it * 128 + t;
            int row = idx >> 5;
            int c4  = (idx & 31) * 4;
            float4 v = *(const float4*)(B + (long long)(kb * 32 + row) * DIM + c4);
            lB[row][c4 + 0] = v.x; lB[row][c4 + 1] = v.y;
            lB[row][c4 + 2] = v.z; lB[row][c4 + 3] = v.w;
        }
        __syncthreads();
        #pragma unroll
        for (int kk = 0; kk < 32; kk += 4) {
            v2f a;
            a.x = lA[wv * 16 + l15][kb * 32 + kk + half * 2];
            a.y = lA[wv * 16 + l15][kb * 32 + kk + half * 2 + 1];
            #pragma unroll
            for (int ct = 0; ct < 8; ++ct) {
                v2f b;
                b.x = lB[kk + half * 2][ct * 16 + l15];
                b.y = lB[kk + half * 2 + 1][ct * 16 + l15];
                acc[ct] = __builtin_amdgcn_wmma_f32_16x16x4_f32(
                    false, a, false, b, (short)0, acc[ct], false, false);
            }
        }
    }

    // ---- epilogue: C layout (VGPR r -> row r for lanes 0-15, row r+8 for lanes 16-31)
    #pragma unroll
    for (int ct = 0; ct < 8; ++ct) {
        int col = ct * 16 + l15;
        float bv = bias[col];
        #pragma unroll
        for (int r = 0; r < 8; ++r) {
            int grow = rowbase + wv * 16 + half * 8 + r;
            if (grow < nrows) C[(long long)grow * DIM + col] = acc[ct][r] + bv;
        }
    }
}

// ---------------------------------------------------------------- per-column sum / sum-of-squares
__global__ void gin_colstats(const float* __restrict__ z, float* __restrict__ sums,
                             float* __restrict__ sumsq, int nrows) {
    int col = threadIdx.x;  // 128 threads
    float s = 0.f, q = 0.f;
    for (int row = blockIdx.x; row < nrows; row += gridDim.x) {
        float v = z[(long long)row * DIM + col];
        s += v; q += v * v;
    }
    atomicAdd(&sums[col], s);
    atomicAdd(&sumsq[col], q);
}

__global__ void gin_bn_finalize(const float* __restrict__ sums, const float* __restrict__ sumsq,
                                const float* __restrict__ g, const float* __restrict__ b,
                                float* __restrict__ scale, float* __restrict__ shift, int nrows) {
    int c = threadIdx.x;
    float inv = 1.0f / (float)nrows;
    float m   = sums[c] * inv;
    float var = sumsq[c] * inv - m * m;      // biased variance, matches reference
    float sc  = g[c] * rsqrtf(var + 1e-5f);
    scale[c] = sc;
    shift[c] = b[c] - m * sc;
}

__global__ void gin_affine_relu(float* __restrict__ z, const float* __restrict__ scale,
                                const float* __restrict__ shift, long long n) {
    long long i4 = ((long long)blockIdx.x * blockDim.x + threadIdx.x) * 4;
    if (i4 >= n) return;
    float4 v = *(float4*)(z + i4);
    int c = (int)(i4 & (DIM - 1));
    v.x = fmaxf(v.x * scale[c + 0] + shift[c + 0], 0.f);
    v.y = fmaxf(v.y * scale[c + 1] + shift[c + 1], 0.f);
    v.z = fmaxf(v.z * scale[c + 2] + shift[c + 2], 0.f);
    v.w = fmaxf(v.w * scale[c + 3] + shift[c + 3], 0.f);
    *(float4*)(z + i4) = v;
}

// ---------------------------------------------------------------- fold fc GEMM into a matvec:
// v_i = fcW[i] @ lastW[(i+1)*128 : (i+2)*128],  c_i = fcb[i] . lastW_slice
__global__ void gin_fold_fc(const float* __restrict__ fcW, const float* __restrict__ fcb,
                            const float* __restrict__ lastW, float* __restrict__ vvec,
                            float* __restrict__ cvec) {
    int i = blockIdx.x;        // layer
    int d = threadIdx.x;       // 0..127
    const float* w = lastW + (i + 1) * DIM;
    const float* Wrow = fcW + ((long long)i * DIM + d) * DIM;
    float s = 0.f;
    for (int j = 0; j < DIM; ++j) s += Wrow[j] * w[j];
    vvec[i * DIM + d] = s;
    if (d == 0) {
        float cc = 0.f;
        for (int j = 0; j < DIM; ++j) cc += fcb[i * DIM + j] * w[j];
        cvec[i] = cc;
    }
}

// logits[n] += h_final[n].lastW[0:128]   (direct, per-node part)
// logits[batch[n]] += sum_i ( outs_i[n].v_i + c_i )   (pooled parts, scatter)
__global__ void gin_node_logits(const float* __restrict__ outs, long long ND,
                                const float* __restrict__ vvec, const float* __restrict__ cvec,
                                const float* __restrict__ lastW, const int* __restrict__ batch,
                                float* __restrict__ logits, int nrows) {
    __shared__ float sa[128];
    __shared__ float sb[128];
    int n = blockIdx.x;
    int t = threadIdx.x;
    const float* hf = outs + 4 * ND;                 // h_final = outs[4]
    float a = hf[(long long)n * DIM + t] * lastW[t];
    float b = 0.f;
    #pragma unroll
    for (int i = 0; i < NLAYERS; ++i)
        b += outs[i * ND + (long long)n * DIM + t] * vvec[i * DIM + t];
    sa[t] = a; sb[t] = b;
    __syncthreads();
    for (int s2 = 64; s2 > 0; s2 >>= 1) {
        if (t < s2) { sa[t] += sa[t + s2]; sb[t] += sb[t + s2]; }
        __syncthreads();
    }
    if (t == 0) {
        float cs = cvec[0] + cvec[1] + cvec[2] + cvec[3] + cvec[4];
        atomicAdd(&logits[n], sa[0]);
        atomicAdd(&logits[batch[n]], sb[0] + cs);
    }
}

__global__ void gin_sigmoid(const float* __restrict__ logits, const float* __restrict__ lastb,
                            float* __restrict__ out, int nrows) {
    int n = blockIdx.x * blockDim.x + threadIdx.x;
    if (n >= nrows) return;
    float L = logits[n] + lastb[0];
    out[n] = 1.f / (1.f + __expf(-L));
    out[nrows + n] = L;
}

// ----------------------------------------------------------------
extern "C" void kernel_launch(void* const* d_in, const int* in_sizes, int n_in,
                              void* d_out, int out_size, void* d_ws, size_t ws_size,
                              hipStream_t stream) {
    const float* x      = (const float*)d_in[0];
    const int*   eidx   = (const int*)d_in[1];
    const int*   batch  = (const int*)d_in[2];
    const float* convW1 = (const float*)d_in[3];
    const float* convb1 = (const float*)d_in[4];
    const float* bn1_g  = (const float*)d_in[5];
    const float* bn1_b  = (const float*)d_in[6];
    const float* convW2 = (const float*)d_in[7];
    const float* convb2 = (const float*)d_in[8];
    const float* bn2_g  = (const float*)d_in[9];
    const float* bn2_b  = (const float*)d_in[10];
    const float* fcW    = (const float*)d_in[11];
    const float* fcb    = (const float*)d_in[12];
    const float* lastW  = (const float*)d_in[13];
    const float* lastb  = (const float*)d_in[14];

    const int N = in_sizes[2];
    const int E = in_sizes[1] / 2;
    const int* src = eidx;
    const int* dst = eidx + E;

    const long long ND = (long long)N * DIM;
    float* ws     = (float*)d_ws;
    float* agg    = ws;                 // ND
    float* z1     = agg + ND;           // ND
    float* outs   = z1 + ND;            // 5*ND
    float* stats  = outs + 5 * ND;      // sums(128) sumsq(128) scale(128) shift(128)
    float* sums   = stats;
    float* sumsq  = stats + 128;
    float* scale  = stats + 256;
    float* shift  = stats + 384;
    float* vvec   = stats + 512;        // 5*128
    float* cvec   = vvec + 5 * DIM;     // 8 (padded)
    float* logits = cvec + 8;           // N

    const int gemmBlocks = (N + 63) / 64;
    const int scatBlocks = (int)(((long long)E * 32 + 255) / 256);

    const float* hcur = x;
    for (int i = 0; i < NLAYERS; ++i) {
        gin_zero_f32<<<4096, 256, 0, stream>>>(agg, ND);
        gin_zero_f32<<<1, 256, 0, stream>>>(sums, 256);   // sums + sumsq
        gin_scatter_agg<<<scatBlocks, 256, 0, stream>>>(hcur, src, dst, agg, E);

        float* outi = outs + (long long)i * ND;
        // z1 = (h + agg) @ W1 + b1
        gin_gemm128<<<gemmBlocks, 128, 0, stream>>>(
            hcur, agg, nullptr, nullptr,
            convW1 + (long long)i * DIM * DIM, convb1 + i * DIM, z1, N, 0);
        gin_colstats<<<1024, 128, 0, stream>>>(z1, sums, sumsq, N);
        gin_bn_finalize<<<1, 128, 0, stream>>>(sums, sumsq, bn1_g + i * DIM, bn1_b + i * DIM,
                                               scale, shift, N);
        gin_zero_f32<<<1, 256, 0, stream>>>(sums, 256);
        // outi = relu(bn(z1)) @ W2 + b2   (BN+ReLU fused into A-staging)
        gin_gemm128<<<gemmBlocks, 128, 0, stream>>>(
            z1, nullptr, scale, shift,
            convW2 + (long long)i * DIM * DIM, convb2 + i * DIM, outi, N, 1);
        gin_colstats<<<1024, 128, 0, stream>>>(outi, sums, sumsq, N);
        gin_bn_finalize<<<1, 128, 0, stream>>>(sums, sumsq, bn2_g + i * DIM, bn2_b + i * DIM,
                                               scale, shift, N);
        gin_affine_relu<<<(int)((ND / 4 + 255) / 256), 256, 0, stream>>>(outi, scale, shift, ND);
        hcur = outi;
    }

    gin_fold_fc<<<NLAYERS, 128, 0, stream>>>(fcW, fcb, lastW, vvec, cvec);
    gin_zero_f32<<<1024, 256, 0, stream>>>(logits, N);
    gin_node_logits<<<N, 128, 0, stream>>>(outs, ND, vvec, cvec, lastW, batch, logits, N);
    gin_sigmoid<<<(N + 255) / 256, 256, 0, stream>>>(logits, lastb, (float*)d_out, N);
}